// DisGAT_63969242907087
// MI455X (gfx1250) — compile-verified
//
#include <hip/hip_runtime.h>
#include <hip/hip_bf16.h>
#include <math.h>

#define C_   4
#define N_   50000
#define E_   500000
#define D_   128
#define DECH 128
#define DECO 64

typedef __attribute__((ext_vector_type(2))) float v2f;
typedef __attribute__((ext_vector_type(8))) float v8f;

// Stage a contiguous 16 x K fp32 A-tile from global into LDS with a padded
// row stride (ldsm = K+4 floats) so WMMA A-operand reads are bank-conflict
// free: dword bank = (r*(K+4) + k) % 64 = (4r + k) % 64 for K % 64 == 0.
// Row byte stride (K+4)*4 is a multiple of 16 -> b128 staging stores legal.
__device__ __forceinline__ void stage_a_tile(float* sm, int ldsm,
                                             const float* __restrict__ g, int K,
                                             int tid, int nth) {
  const int cpr   = K / 4;          // float4 chunks per row
  const int total = 16 * cpr;
  for (int i = tid; i < total; i += nth) {
    const int rr = i / cpr;
    const int cc = (i - rr * cpr) * 4;
    *(float4*)(sm + (size_t)rr * ldsm + cc) =
        *(const float4*)(g + (size_t)rr * K + cc);
  }
  __syncthreads();
}

// One wave computes a 16(M) x 16(N) fp32 tile with V_WMMA_F32_16X16X4_F32,
// A read from LDS (row stride ldsm), B from global (L2-resident weights).
// A operand layout (ISA 7.12.2): lanes 0-15 hold K={0,1}, lanes 16-31 K={2,3},
// row M = lane&15. B operand: rows K striped per half-wave, column N = lane&15.
__device__ __forceinline__ v8f wmma_rowtile(const float* As, int ldsm,
                                            const float* __restrict__ B, int ldb,
                                            int K, int lane) {
  const int half = lane >> 4;
  const int r    = lane & 15;
  const float* Arow = As + (size_t)r * ldsm;
  v8f acc = {};
  for (int k = 0; k < K; k += 4) {
    const int ka = k + 2 * half;
    v2f a = *(const v2f*)(Arow + ka);                 // ds_load_b64
    v2f b;
    b.x = B[(size_t)ka       * ldb + r];
    b.y = B[(size_t)(ka + 1) * ldb + r];
    acc = __builtin_amdgcn_wmma_f32_16x16x4_f32(false, a, false, b,
                                                (short)0, acc, false, false);
  }
  return acc;
}

// Out[c,n,:] = (In[c,n,:] @ W[c] + Bias[c,:]) * scale ; optional copy to Out2.
// grid = (N/16, C), block = 256 (8 waves = 8 column tiles of D=128).
__global__ void gemm_cnd(const float* __restrict__ In, const float* __restrict__ W,
                         const float* __restrict__ Bias, float* __restrict__ Out,
                         float* __restrict__ Out2, float scale) {
  __shared__ float smA[16 * (D_ + 4)];
  const int lane = threadIdx.x & 31;
  const int wid  = threadIdx.x >> 5;    // column tile 0..7
  const int m0   = blockIdx.x * 16;
  const int c    = blockIdx.y;

  const float* A  = In + ((size_t)c * N_ + m0) * D_;   // 16x128, contiguous
  const float* Bp = W + (size_t)c * D_ * D_ + wid * 16;
  __builtin_prefetch(Bp, 0, 1);                        // global_prefetch_b8

  stage_a_tile(smA, D_ + 4, A, D_, threadIdx.x, blockDim.x);
  v8f acc = wmma_rowtile(smA, D_ + 4, Bp, D_, D_, lane);

  const int   col  = wid * 16 + (lane & 15);
  const int   half = lane >> 4;
  const float bv   = Bias[c * D_ + col];

  float* O  = Out + ((size_t)c * N_ + m0) * D_ + col;
  float* O2 = Out2 ? Out2 + ((size_t)c * N_ + m0) * D_ + col : nullptr;
#pragma unroll
  for (int i = 0; i < 8; ++i) {
    // C/D layout: VGPR i holds rows M=i (lanes 0-15) and M=i+8 (lanes 16-31)
    const float v = (acc[i] + bv) * scale;
    O[(size_t)(i + 8 * half) * D_] = v;
    if (O2) O2[(size_t)(i + 8 * half) * D_] = v;
  }
}

// One wave per edge. scores[k][c] = <Hn[c,src,:], Alp[k,dst,:]>; leaky-relu;
// softmax over c; p[k]=w[k][k]; Hout[k,dst,:] += p[k]*Hn[k,src,:] (atomics).
// Hout pre-initialized to Hn (gemm_cnd Out2 path) -> result is Hn + agg.
// Wout (C,C,E) written only for the last layer.
__global__ void edge_kernel(const float* __restrict__ Hn, const float* __restrict__ Alp,
                            const long long* __restrict__ ei, float* __restrict__ Hout,
                            float* __restrict__ Wout) {
  const int lane = threadIdx.x & 31;
  const int wid  = threadIdx.x >> 5;
  const size_t e = (size_t)blockIdx.x * 8 + wid;
  if (e >= E_) return;

  const int src = (int)ei[e];
  const int dst = (int)ei[(size_t)E_ + e];

  float4 xj[C_], ai[C_];
#pragma unroll
  for (int c = 0; c < C_; ++c) {
    xj[c] = *(const float4*)(Hn  + ((size_t)c * N_ + src) * D_ + lane * 4);
    ai[c] = *(const float4*)(Alp + ((size_t)c * N_ + dst) * D_ + lane * 4);
  }

  float w[C_][C_];
#pragma unroll
  for (int k = 0; k < C_; ++k) {
#pragma unroll
    for (int c = 0; c < C_; ++c) {
      float s = xj[c].x * ai[k].x + xj[c].y * ai[k].y +
                xj[c].z * ai[k].z + xj[c].w * ai[k].w;
#pragma unroll
      for (int off = 16; off > 0; off >>= 1)  // butterfly: all lanes get total
        s += __shfl_xor(s, off, 32);
      w[k][c] = s >= 0.0f ? s : 0.01f * s;    // leaky relu
    }
  }

#pragma unroll
  for (int k = 0; k < C_; ++k) {              // softmax over c
    float m = w[k][0];
#pragma unroll
    for (int c = 1; c < C_; ++c) m = fmaxf(m, w[k][c]);
    float sum = 0.0f;
#pragma unroll
    for (int c = 0; c < C_; ++c) { w[k][c] = __expf(w[k][c] - m); sum += w[k][c]; }
    const float inv = 1.0f / sum;
#pragma unroll
    for (int c = 0; c < C_; ++c) w[k][c] *= inv;
  }

  if (Wout != nullptr && lane == 0) {
#pragma unroll
    for (int k = 0; k < C_; ++k)
#pragma unroll
      for (int c = 0; c < C_; ++c)
        Wout[(size_t)(k * C_ + c) * E_ + e] = w[k][c];
  }

#pragma unroll
  for (int k = 0; k < C_; ++k) {
    const float p = w[k][k];
    float* dp = Hout + ((size_t)k * N_ + dst) * D_ + lane * 4;
    unsafeAtomicAdd(dp + 0, p * xj[k].x);
    unsafeAtomicAdd(dp + 1, p * xj[k].y);
    unsafeAtomicAdd(dp + 2, p * xj[k].z);
    unsafeAtomicAdd(dp + 3, p * xj[k].w);
  }
}

// Hc[n, k*D+d] = Hf[k, n, d]   (both sides coalesced over d)
__global__ void concat_kernel(const float* __restrict__ Hf, float* __restrict__ Hc) {
  const size_t idx = (size_t)blockIdx.x * blockDim.x + threadIdx.x;
  const size_t total = (size_t)N_ * C_ * D_;
  if (idx >= total) return;
  const int    d = (int)(idx % D_);
  const int    k = (int)((idx / D_) % C_);
  const size_t n = idx / ((size_t)D_ * C_);
  Hc[idx] = Hf[((size_t)k * N_ + n) * D_ + d];
}

// relu(In @ W + b) for two decoders (blockIdx.y selects weight set).
// block = (Nout/16) waves * 32.
__global__ void gemm_dec(const float* __restrict__ In0, const float* __restrict__ In1,
                         int K, int Nout,
                         const float* __restrict__ W0, const float* __restrict__ B0,
                         const float* __restrict__ W1, const float* __restrict__ B1,
                         float* __restrict__ Out0, float* __restrict__ Out1) {
  __shared__ float smA[16 * (512 + 4)];   // max K = C*D = 512
  const int lane = threadIdx.x & 31;
  const int wid  = threadIdx.x >> 5;
  const int m0   = blockIdx.x * 16;
  const int dec  = blockIdx.y;

  const float* In = dec ? In1 : In0;
  const float* W  = dec ? W1  : W0;
  const float* Bb = dec ? B1  : B0;
  float*       Ou = dec ? Out1 : Out0;

  const float* Bp = W + wid * 16;
  __builtin_prefetch(Bp, 0, 1);

  stage_a_tile(smA, K + 4, In + (size_t)m0 * K, K, threadIdx.x, blockDim.x);
  v8f acc = wmma_rowtile(smA, K + 4, Bp, Nout, K, lane);

  const int   col  = wid * 16 + (lane & 15);
  const int   half = lane >> 4;
  const float bv   = Bb[col];
  float* O = Ou + (size_t)m0 * Nout + col;
#pragma unroll
  for (int i = 0; i < 8; ++i)
    O[(size_t)(i + 8 * half) * Nout] = fmaxf(acc[i] + bv, 0.0f);
}

extern "C" void kernel_launch(void* const* d_in, const int* in_sizes, int n_in,
                              void* d_out, int out_size, void* d_ws, size_t ws_size,
                              hipStream_t stream) {
  const float*     H     = (const float*)d_in[0];
  const long long* ei    = (const long long*)d_in[1];   // int64 edge_index (2,E)
  const float*     lin_w = (const float*)d_in[2];       // (L,C,D,D)
  const float*     lin_b = (const float*)d_in[3];       // (L,C,D)
  const float*     q_w   = (const float*)d_in[4];
  const float*     q_b   = (const float*)d_in[5];
  const float*     d2w1  = (const float*)d_in[6];
  const float*     d2b1  = (const float*)d_in[7];
  const float*     d2w2  = (const float*)d_in[8];
  const float*     d2b2  = (const float*)d_in[9];
  const float*     d3w1  = (const float*)d_in[10];
  const float*     d3b1  = (const float*)d_in[11];
  const float*     d3w2  = (const float*)d_in[12];
  const float*     d3b2  = (const float*)d_in[13];

  float* out = (float*)d_out;
  float* ws  = (float*)d_ws;

  const size_t CND = (size_t)C_ * N_ * D_;     // 25.6M floats
  float* Hn  = ws;
  float* Alp = ws + CND;
  float* HA  = ws + 2 * CND;
  float* HB  = ws + 3 * CND;
  float* hid = ws + 4 * CND;                   // 2 * N * DECH floats

  float* outHc = out;                                      // (N, C*D)
  float* outW  = out + (size_t)N_ * C_ * D_;               // (C, C, E)
  float* outD2 = outW + (size_t)C_ * C_ * E_;              // (N, DECO)
  float* outD3 = outD2 + (size_t)N_ * DECO;                // (N, DECO)

  const float inv_sqrt_d = 1.0f / sqrtf((float)D_);
  const size_t WL = (size_t)C_ * D_ * D_;  // per-layer weight stride
  const size_t BL = (size_t)C_ * D_;       // per-layer bias stride

  const dim3 gg(N_ / 16, C_);

  // ---- layer 0 ----
  gemm_cnd<<<gg, 256, 0, stream>>>(H, lin_w, lin_b, Hn, HA, 1.0f);
  gemm_cnd<<<gg, 256, 0, stream>>>(Hn, q_w, q_b, Alp, nullptr, inv_sqrt_d);
  edge_kernel<<<E_ / 8, 256, 0, stream>>>(Hn, Alp, ei, HA, nullptr);

  // ---- layer 1 ----
  gemm_cnd<<<gg, 256, 0, stream>>>(HA, lin_w + WL, lin_b + BL, Hn, HB, 1.0f);
  gemm_cnd<<<gg, 256, 0, stream>>>(Hn, q_w + WL, q_b + BL, Alp, nullptr, inv_sqrt_d);
  edge_kernel<<<E_ / 8, 256, 0, stream>>>(Hn, Alp, ei, HB, outW);

  // ---- concat ----
  const size_t tot = CND;
  concat_kernel<<<(unsigned)((tot + 255) / 256), 256, 0, stream>>>(HB, outHc);

  // ---- decoders ----
  float* hid2 = hid;
  float* hid3 = hid + (size_t)N_ * DECH;
  gemm_dec<<<dim3(N_ / 16, 2), 256, 0, stream>>>(outHc, outHc, C_ * D_, DECH,
                                                 d2w1, d2b1, d3w1, d3b1, hid2, hid3);
  gemm_dec<<<dim3(N_ / 16, 2), 128, 0, stream>>>(hid2, hid3, DECH, DECO,
                                                 d2w2, d2b2, d3w2, d3b2, outD2, outD3);
}